// TrafficQuantumLayer_45414984188040
// MI455X (gfx1250) — compile-verified
//
#include <hip/hip_runtime.h>
#include <hip/hip_bf16.h>
#include <math.h>

// Shapes from the reference
#define N_QUBITS 4
#define N_LAYERS 2
#define IN_FEAT  8
#define OUT_FEAT 128
#define BATCH    262144

typedef __attribute__((ext_vector_type(2))) float v2f;
typedef __attribute__((ext_vector_type(8))) float v8f;

// RY(theta) on the wire whose bit value is `bit` (wire w -> bit 8>>w):
// for pairs (i, i|bit): a' = c*a - s*b ; b' = s*a + c*b
__device__ __forceinline__ void apply_ry(float st[16], float c, float s, int bit) {
#pragma unroll
  for (int i = 0; i < 16; ++i) {
    if (i & bit) continue;
    const float a = st[i];
    const float b = st[i | bit];
    st[i]       = c * a - s * b;
    st[i | bit] = s * a + c * b;
  }
}

// CNOT(ctrl,tgt): for states with ctrl bit set, flip tgt bit (swap amplitudes)
__device__ __forceinline__ void apply_cnot(float st[16], int cbit, int tbit) {
#pragma unroll
  for (int i = 0; i < 16; ++i) {
    if ((i & cbit) && !(i & tbit)) {
      const float t  = st[i];
      st[i]          = st[i | tbit];
      st[i | tbit]   = t;
    }
  }
}

__global__ __launch_bounds__(256) void tq_fused_kernel(
    const float* __restrict__ x,       // [B, 8]
    const float* __restrict__ W_pre,   // [4, 8]
    const float* __restrict__ b_pre,   // [4]
    const float* __restrict__ theta,   // [2, 4]
    const float* __restrict__ W_post,  // [128, 4]
    const float* __restrict__ b_post,  // [128]
    float* __restrict__ out)           // [B, 128]
{
  const int lane    = threadIdx.x & 31;
  const int waveId  = blockIdx.x * 8 + (threadIdx.x >> 5);
  const int rowBase = waveId * 16;          // 16 batch rows per wave
  const int sub     = lane & 15;
  const bool hi     = lane >= 16;
  const int myRow   = rowBase + sub;        // lanes L and L+16 share a row

  // ---- load x row (32B, 16B-aligned vector loads) ----
  const float4 x0 = *(const float4*)(x + (size_t)myRow * IN_FEAT);
  const float4 x1 = *(const float4*)(x + (size_t)myRow * IN_FEAT + 4);
  const float xr[8] = {x0.x, x0.y, x0.z, x0.w, x1.x, x1.y, x1.z, x1.w};

  // ---- alpha = pi * tanh(x @ W_pre^T + b_pre) ----
  float alpha[N_QUBITS];
#pragma unroll
  for (int q = 0; q < N_QUBITS; ++q) {
    float acc = b_pre[q];
#pragma unroll
    for (int k = 0; k < IN_FEAT; ++k)
      acc = fmaf(xr[k], W_pre[q * IN_FEAT + k], acc);
    alpha[q] = 3.14159265358979323846f * tanhf(acc);
  }

  // ---- 4-qubit real statevector circuit in registers ----
  float st[16];
#pragma unroll
  for (int i = 0; i < 16; ++i) st[i] = 0.0f;
  st[0] = 1.0f;

  // Embedding RYs (wire w -> bit value 8>>w)
#pragma unroll
  for (int w = 0; w < N_QUBITS; ++w) {
    float s, c;
    __sincosf(alpha[w] * 0.5f, &s, &c);
    apply_ry(st, c, s, 8 >> w);
  }
  // Trainable layers: RY(theta[l,w]) then CNOT chain 0-1,1-2,2-3
#pragma unroll
  for (int l = 0; l < N_LAYERS; ++l) {
#pragma unroll
    for (int w = 0; w < N_QUBITS; ++w) {
      float s, c;
      __sincosf(theta[l * N_QUBITS + w] * 0.5f, &s, &c);
      apply_ry(st, c, s, 8 >> w);
    }
#pragma unroll
    for (int w = 0; w < N_QUBITS - 1; ++w)
      apply_cnot(st, 8 >> w, 8 >> (w + 1));
  }

  // ---- PauliZ expectations z[w] = sum_i (+-) st[i]^2 ----
  float z[N_QUBITS] = {0.0f, 0.0f, 0.0f, 0.0f};
#pragma unroll
  for (int i = 0; i < 16; ++i) {
    const float p = st[i] * st[i];
#pragma unroll
    for (int w = 0; w < N_QUBITS; ++w)
      z[w] += (i & (8 >> w)) ? -p : p;
  }

  // ---- A fragment: 16x4 f32 per ISA layout ----
  // lanes 0-15: {A[M][0], A[M][1]}  lanes 16-31: {A[M][2], A[M][3]}, M = lane&15
  v2f afrag;
  afrag.x = hi ? z[2] : z[0];
  afrag.y = hi ? z[3] : z[1];

  const int kBase = hi ? 2 : 0;
  const int mBase = rowBase + (hi ? 8 : 0);

  // ---- 8 x V_WMMA_F32_16X16X4_F32 over the 128 output columns ----
#pragma unroll
  for (int ob = 0; ob < 8; ++ob) {
    const int o = ob * 16 + sub;  // this lane's output column in the block

    // B fragment: B[k][n] = W_post[o][k]; lane holds {B[kBase][n], B[kBase+1][n]}
    const float2 wp = *(const float2*)(W_post + o * N_QUBITS + kBase);
    v2f bfrag;
    bfrag.x = wp.x;
    bfrag.y = wp.y;

    // C fragment: bias broadcast down the column (all 8 row-VGPRs same value)
    const float bias = b_post[o];
    v8f cfrag;
#pragma unroll
    for (int j = 0; j < 8; ++j) cfrag[j] = bias;

    // D = A x B + C   (8 args: neg_a, A, neg_b, B, c_mod, C, reuse_a, reuse_b)
    v8f dfrag = __builtin_amdgcn_wmma_f32_16x16x4_f32(
        false, afrag, false, bfrag, (short)0, cfrag, false, false);

    // D VGPR j: lanes 0-15 -> row rowBase+j, lanes 16-31 -> row rowBase+8+j
#pragma unroll
    for (int j = 0; j < 8; ++j) {
      __builtin_nontemporal_store(
          dfrag[j], out + (size_t)(mBase + j) * OUT_FEAT + ob * 16 + sub);
    }
  }
}

extern "C" void kernel_launch(void* const* d_in, const int* in_sizes, int n_in,
                              void* d_out, int out_size, void* d_ws, size_t ws_size,
                              hipStream_t stream) {
  const float* x      = (const float*)d_in[0];
  const float* W_pre  = (const float*)d_in[1];
  const float* b_pre  = (const float*)d_in[2];
  const float* theta  = (const float*)d_in[3];
  const float* W_post = (const float*)d_in[4];
  const float* b_post = (const float*)d_in[5];
  float* out = (float*)d_out;

  // 16 rows per wave, 8 waves per block -> 128 rows per block
  const int blocks = BATCH / 128;  // 2048
  tq_fused_kernel<<<blocks, 256, 0, stream>>>(x, W_pre, b_pre, theta,
                                              W_post, b_post, out);
}